// HarmonicSynth_37374805410530
// MI455X (gfx1250) — compile-verified
//
#include <hip/hip_runtime.h>
#include <math.h>

// ---------------------------------------------------------------------------
// DDSP HarmonicSynth on MI455X (gfx1250, wave32)
//
// out[b, f*256 + t] = sum_{h=1..64} sin(h * omega_t) * w[b,f,h]
//   w = harmonic_distribution * amplitudes      (constant within a frame)
//   omega_t = base_f + (t+1)*Delta_f            (affine in t within a frame)
//
// With t = m + 16n  ->  omega = alpha_m + beta_n, and
//   sin(h(alpha+beta)) = sin(h a)cos(h b) + cos(h a)sin(h b)
// the 256-sample frame tile is  O = A1*B1 + A2*B2  (two 16x16x64 f32 GEMMs),
// executed as 32 chained V_WMMA_F32_16X16X4_F32 per wave (one wave per frame).
// ---------------------------------------------------------------------------

typedef __attribute__((ext_vector_type(2))) float v2f;
typedef __attribute__((ext_vector_type(8))) float v8f;

#define TWO_PI   6.283185307179586f
#define INV_2PI  0.15915494309189535f
#define SR       16000.0f

// ---- Kernel 1: per-batch exclusive prefix of per-frame phase advance -------
// base[b][f] = sum_{f' < f} 256 * (2*pi*f0[b][f']/SR)
__global__ __launch_bounds__(256) void phase_scan_kernel(
    const float* __restrict__ f0, float* __restrict__ base, int frames)
{
    __shared__ float buf[256];
    const int b   = blockIdx.x;
    const int tid = threadIdx.x;
    float running = 0.0f;

    for (int c0 = 0; c0 < frames; c0 += 256) {
        const int idx = c0 + tid;
        float v = 0.0f;
        if (idx < frames)
            v = 256.0f * (TWO_PI * f0[b * frames + idx] * (1.0f / SR));
        buf[tid] = v;
        __syncthreads();
        // Hillis-Steele inclusive scan over 256 threads
        #pragma unroll
        for (int off = 1; off < 256; off <<= 1) {
            float add = (tid >= off) ? buf[tid - off] : 0.0f;
            __syncthreads();
            buf[tid] += add;
            __syncthreads();
        }
        const float incl  = buf[tid];
        const float total = buf[255];
        if (idx < frames)
            base[b * frames + idx] = running + (incl - v);  // exclusive
        running += total;
        __syncthreads();  // protect buf before next chunk overwrites it
    }
}

// ---- Kernel 2: one wave32 per frame, 32x V_WMMA_F32_16X16X4_F32 ------------
__global__ __launch_bounds__(32) void harmonic_wmma_kernel(
    const float* __restrict__ amp,      // (B*F)
    const float* __restrict__ hd,       // (B*F, 64)
    const float* __restrict__ f0,       // (B*F)
    const float* __restrict__ base_arr, // (B*F) from phase_scan_kernel
    float* __restrict__ out)            // (B*F, 256)
{
    __shared__ float ws[64];    // harmonic weights for this frame
    __shared__ float so[256];   // output staging for coalesced stores

    const int bf    = blockIdx.x;   // b*frames + f
    const int l     = threadIdx.x;  // 0..31
    const int idx16 = l & 15;       // M (A role) / N (B role)
    const int hi    = l >> 4;       // lane-half selects K pair: k = 2*hi + j

    // weights: w[h] = hd[h] * amp  (h index 0..63 <-> harmonic 1..64)
    const float a = amp[bf];
    ws[l]      = hd[bf * 64 + l]      * a;
    ws[l + 32] = hd[bf * 64 + l + 32] * a;
    __syncthreads();

    const float delta = TWO_PI * f0[bf] * (1.0f / SR);
    float base = base_arr[bf];
    base -= TWO_PI * truncf(base * INV_2PI);  // range-reduce for accuracy

    const float alpha = base + (float)(idx16 + 1) * delta;  // row angle
    const float beta  = (float)idx16 * 16.0f * delta;       // col angle

    // Init (sin,cos) of (k+1)*alpha and (k+1)*beta for this lane's two k
    // slots, plus the e^{i*4*angle} rotation step shared by both slots.
    float sA[2], cA[2], sB[2], cB[2];
    #pragma unroll
    for (int j = 0; j < 2; ++j) {
        const int k = 2 * hi + j;
        sincosf((float)(k + 1) * alpha, &sA[j], &cA[j]);
        sincosf((float)(k + 1) * beta,  &sB[j], &cB[j]);
    }
    float S4A, C4A, S4B, C4B;
    sincosf(4.0f * alpha, &S4A, &C4A);
    sincosf(4.0f * beta,  &S4B, &C4B);

    v8f acc = {0.f, 0.f, 0.f, 0.f, 0.f, 0.f, 0.f, 0.f};

    // 16 K-chunks of 4 harmonics each: h = 4c + k + 1
    #pragma unroll
    for (int c = 0; c < 16; ++c) {
        v2f a1, b1, a2, b2;
        #pragma unroll
        for (int j = 0; j < 2; ++j) {
            const int k = 2 * hi + j;
            const float w = ws[4 * c + k];     // w_{h}, h = 4c+k+1
            a1[j] = w * sA[j];                 // w_h * sin(h*alpha_m)
            a2[j] = w * cA[j];                 // w_h * cos(h*alpha_m)
            b1[j] = cB[j];                     // cos(h*beta_n)
            b2[j] = sB[j];                     // sin(h*beta_n)
            // rotate to h+4 for the next chunk
            float s = sA[j], cc = cA[j];
            sA[j] = s * C4A + cc * S4A;
            cA[j] = cc * C4A - s * S4A;
            s = sB[j]; cc = cB[j];
            sB[j] = s * C4B + cc * S4B;
            cB[j] = cc * C4B - s * S4B;
        }
        acc = __builtin_amdgcn_wmma_f32_16x16x4_f32(
            false, a1, false, b1, (short)0, acc, false, false);
        acc = __builtin_amdgcn_wmma_f32_16x16x4_f32(
            false, a2, false, b2, (short)0, acc, false, false);
    }

    // C/D layout: vgpr r, lane l -> M = r + 8*(l/16), N = l%16; t = M + 16N
    #pragma unroll
    for (int r = 0; r < 8; ++r) {
        const int t = (r + 8 * hi) + 16 * idx16;
        so[t] = acc[r];
    }
    __syncthreads();

    // coalesced b128 stores
    float4* op = (float4*)(out + (size_t)bf * 256);
    const float4* sp = (const float4*)so;
    op[l]      = sp[l];
    op[l + 32] = sp[l + 32];
}

// ---------------------------------------------------------------------------
extern "C" void kernel_launch(void* const* d_in, const int* in_sizes, int n_in,
                              void* d_out, int out_size, void* d_ws, size_t ws_size,
                              hipStream_t stream) {
    const float* amp = (const float*)d_in[0];  // (B, F, 1)
    const float* hd  = (const float*)d_in[1];  // (B, F, 64)
    const float* f0  = (const float*)d_in[2];  // (B, F, 1)
    float* out = (float*)d_out;

    const int n_bf   = in_sizes[0];   // B * FRAMES = 4000
    const int B      = 4;
    const int frames = n_bf / B;      // 1000

    float* base = (float*)d_ws;       // n_bf floats of scratch

    phase_scan_kernel<<<B, 256, 0, stream>>>(f0, base, frames);
    harmonic_wmma_kernel<<<n_bf, 32, 0, stream>>>(amp, hd, f0, base, out);
}